// ChamferDistance_1486058684542
// MI455X (gfx1250) — compile-verified
//
#include <hip/hip_runtime.h>
#include <hip/hip_bf16.h>

typedef __attribute__((ext_vector_type(2))) float v2f;
typedef __attribute__((ext_vector_type(8))) float v8f;

#define WAVES_PER_WG    8
#define ROWS_PER_WAVE   16
#define ROWS_PER_WG     (WAVES_PER_WG * ROWS_PER_WAVE)   // 128
#define CHUNK           1024                             // DB points staged in LDS per outer iter
#define TILES_PER_CHUNK (CHUNK / 16)                     // 64
#define GROUP           8                                // tiles per software-pipeline stage
#define NGROUPS         (TILES_PER_CHUNK / GROUP)        // 8

// Load B-fragments for GROUP consecutive 16-point tiles from LDS.
// Per lane: 8 bytes at (t*256 + lane*8); pairs of tiles are 256B apart, so the
// compiler emits ds_load_2addr_b64 with immediate offsets (group span 1792B fits).
__device__ __forceinline__ void load_group(const float* __restrict__ qfrag,
                                           int t0, int lane, v2f b[GROUP]) {
#pragma unroll
    for (int j = 0; j < GROUP; ++j)
        b[j] = *(const v2f*)&qfrag[(t0 + j) * 64 + lane * 2];
}

// 8 WMMAs + min-accumulate (compiler fuses fminf chains into v_min3_num_f32).
__device__ __forceinline__ void compute_group(const v2f afrag, const v2f b[GROUP],
                                              v8f& rmin, const v8f zeroC) {
#pragma unroll
    for (int j = 0; j < GROUP; ++j) {
        v8f d = __builtin_amdgcn_wmma_f32_16x16x4_f32(
            false, afrag, false, b[j], (short)0, zeroC, false, false);
#pragma unroll
        for (int i = 0; i < 8; ++i) rmin[i] = fminf(rmin[i], d[i]);
    }
}

// One-direction nearest-neighbor min-sq-dist:
//   out[b][r] = min_q ( ||P[b][r] - Q[b][q]||^2 ), clamped at 0.
// WMMA K=4 trick: A row i = (x1,y1,z1,1), B col j = (-2x2,-2y2,-2z2,|q|^2)
//   => D[i][j] = |q_j|^2 - 2 p_i.q_j ; add |p_i|^2 and clamp once at the end
// (both commute with the min reduction).
__global__ __launch_bounds__(256) void chamfer_nn_kernel(
    const float* __restrict__ P,   // [B, nP, 3] query points (rows)
    const float* __restrict__ Q,   // [B, nQ, 3] database points (cols)
    float* __restrict__ out,       // [B, nP]
    int nP, int nQ)
{
    __shared__ float qfrag[TILES_PER_CHUNK * 64];  // 16 KB: B-fragments for CHUNK points

    const int stripsPerBatch = nP / ROWS_PER_WG;
    const int b     = blockIdx.x / stripsPerBatch;
    const int strip = blockIdx.x % stripsPerBatch;
    const int tid   = threadIdx.x;
    const int lane  = tid & 31;
    const int wave  = tid >> 5;
    const int n0    = strip * ROWS_PER_WG + wave * ROWS_PER_WAVE;

    const float* __restrict__ Pb = P + (size_t)b * nP * 3;
    const float* __restrict__ Qb = Q + (size_t)b * nQ * 3;

    // ---- A fragment (16x4 f32): lanes 0-15 -> (x,y) = K0,K1 ; lanes 16-31 -> (z,1) = K2,K3
    v2f afrag;
    {
        const int r = n0 + (lane & 15);
        const float x = Pb[r * 3 + 0];
        const float y = Pb[r * 3 + 1];
        const float z = Pb[r * 3 + 2];
        const bool lo = (lane < 16);
        afrag.x = lo ? x : z;
        afrag.y = lo ? y : 1.0f;
    }

    v8f rmin;
#pragma unroll
    for (int i = 0; i < 8; ++i) rmin[i] = 3.0e38f;

    const v8f zeroC = {0.f, 0.f, 0.f, 0.f, 0.f, 0.f, 0.f, 0.f};

    for (int qbase = 0; qbase < nQ; qbase += CHUNK) {
        __syncthreads();  // protect previous chunk's readers

        // ---- cooperative build of B fragments into LDS (4 points per thread)
#pragma unroll
        for (int k = 0; k < CHUNK / 256; ++k) {
            const int p  = tid + k * 256;          // 0..CHUNK-1
            const int gq = qbase + p;
            const float x  = Qb[gq * 3 + 0];
            const float y  = Qb[gq * 3 + 1];
            const float z  = Qb[gq * 3 + 2];
            const float sq = x * x + y * y + z * z;
            const int t = p >> 4;
            const int l = p & 15;
            float* dst = &qfrag[t * 64];
            dst[l * 2 + 0]        = -2.0f * x;     // lane l    (K0)
            dst[l * 2 + 1]        = -2.0f * y;     // lane l    (K1)
            dst[(16 + l) * 2 + 0] = -2.0f * z;     // lane l+16 (K2)
            dst[(16 + l) * 2 + 1] = sq;            // lane l+16 (K3)
        }
        __syncthreads();

        // ---- software-pipelined hot loop: prefetch next group's fragments
        // while the current group's 8 WMMAs + 32 v_min3 execute. Ping-pong
        // buffers avoid register copies; final prefetch wraps to group 0
        // (valid LDS, result discarded) to stay branchless.
        v2f b0[GROUP], b1[GROUP];
        load_group(qfrag, 0, lane, b0);
#pragma unroll 1
        for (int g = 0; g < NGROUPS; g += 2) {
            load_group(qfrag, (g + 1) * GROUP, lane, b1);
            compute_group(afrag, b0, rmin, zeroC);
            load_group(qfrag, ((g + 2) & (NGROUPS - 1)) * GROUP, lane, b0);
            compute_group(afrag, b1, rmin, zeroC);
        }
    }

    // ---- cross-lane row reduction: min over the 16 lanes of each half.
    // C/D layout: (VGPR v, lane-half h) holds row v + 8*h, N = lane & 15.
#pragma unroll
    for (int i = 0; i < 8; ++i) {
        float v = rmin[i];
        v = fminf(v, __shfl_xor(v, 1, 32));
        v = fminf(v, __shfl_xor(v, 2, 32));
        v = fminf(v, __shfl_xor(v, 4, 32));
        v = fminf(v, __shfl_xor(v, 8, 32));
        rmin[i] = v;
    }

    // ---- lanes 0 and 16 commit rows n0+0..7 and n0+8..15 (add ||p||^2, clamp)
    if ((lane & 15) == 0) {
        const int rbase = n0 + (lane >> 4) * 8;
#pragma unroll
        for (int i = 0; i < 8; ++i) {
            const int r = rbase + i;
            const float x = Pb[r * 3 + 0];
            const float y = Pb[r * 3 + 1];
            const float z = Pb[r * 3 + 2];
            const float sq = x * x + y * y + z * z;
            out[(size_t)b * nP + r] = fmaxf(rmin[i] + sq, 0.0f);
        }
    }
}

extern "C" void kernel_launch(void* const* d_in, const int* in_sizes, int n_in,
                              void* d_out, int out_size, void* d_ws, size_t ws_size,
                              hipStream_t stream) {
    (void)n_in; (void)d_ws; (void)ws_size; (void)out_size;

    const float* xyz1 = (const float*)d_in[0];   // [B, N, 3]
    const float* xyz2 = (const float*)d_in[1];   // [B, M, 3]

    const int B = 8;
    const int N = in_sizes[0] / (B * 3);         // 8192
    const int M = in_sizes[1] / (B * 3);         // 8192

    float* dist1 = (float*)d_out;                // [B, N]
    float* dist2 = dist1 + (size_t)B * N;        // [B, M]

    // Pass 1: for each xyz1 point, min sq-dist to xyz2
    chamfer_nn_kernel<<<B * (N / ROWS_PER_WG), 256, 0, stream>>>(xyz1, xyz2, dist1, N, M);
    // Pass 2: roles swapped
    chamfer_nn_kernel<<<B * (M / ROWS_PER_WG), 256, 0, stream>>>(xyz2, xyz1, dist2, M, N);
}